// ListNetRankLoss_19121194402348
// MI455X (gfx1250) — compile-verified
//
#include <hip/hip_runtime.h>

// ListNet masked softmax-CE loss, MI455X (gfx1250).
// Memory-bound: 288 MiB read-once -> ~12.4 us at 23.3 TB/s. Strategy:
// one workgroup per row; async-DMA (ASYNCcnt) the 64KB of logits into LDS
// with non-temporal b128 transfers (stream-through, don't thrash the 192MB L2),
// overlap mask loads, then run both softmax passes from LDS so HBM traffic is
// exactly one pass over the data.

#define N_COLS        8192
#define THREADS       256
#define F4_PER_ROW    (N_COLS / 4)          // 2048 float4 per array per row
#define F4_PER_THREAD (F4_PER_ROW / THREADS) // 8

__device__ __forceinline__ float wave_max(float v) {
#pragma unroll
    for (int o = 16; o > 0; o >>= 1) v = fmaxf(v, __shfl_xor(v, o, 32));
    return v;
}
__device__ __forceinline__ float wave_sum(float v) {
#pragma unroll
    for (int o = 16; o > 0; o >>= 1) v += __shfl_xor(v, o, 32);
    return v;
}

// Fused 3-value reduction across the 8 wave32s of the block; results broadcast
// to all threads. a,b reduced with max or sum (is_max_ab); c always summed.
// sc must be a 24-float scratch region private to this call site.
__device__ __forceinline__ void block_reduce3(float& a, float& b, float& c,
                                              bool is_max_ab, float* sc) {
    if (is_max_ab) { a = wave_max(a); b = wave_max(b); }
    else           { a = wave_sum(a); b = wave_sum(b); }
    c = wave_sum(c);
    const int wid = threadIdx.x >> 5;
    if ((threadIdx.x & 31) == 0) { sc[wid] = a; sc[8 + wid] = b; sc[16 + wid] = c; }
    __syncthreads();
    if (threadIdx.x == 0) {
        float ra = sc[0], rb = sc[8], rc = sc[16];
#pragma unroll
        for (int i = 1; i < 8; ++i) {
            ra = is_max_ab ? fmaxf(ra, sc[i])     : (ra + sc[i]);
            rb = is_max_ab ? fmaxf(rb, sc[8 + i]) : (rb + sc[8 + i]);
            rc += sc[16 + i];
        }
        sc[0] = ra; sc[8] = rb; sc[16] = rc;
    }
    __syncthreads();
    a = sc[0]; b = sc[8]; c = sc[16];
}

__global__ __launch_bounds__(THREADS)
void listnet_row_kernel(const float* __restrict__ pred,
                        const float* __restrict__ targ,
                        const unsigned* __restrict__ mask32,
                        float* __restrict__ ce_out,
                        float* __restrict__ ok_out)
{
    __shared__ float4 sh_t[F4_PER_ROW];   // 32 KB
    __shared__ float4 sh_p[F4_PER_ROW];   // 32 KB
    __shared__ float  sredA[24];          // pass-1 reduction scratch
    __shared__ float  sredB[24];          // pass-2 reduction scratch

    const int    tid   = threadIdx.x;
    const int    row   = blockIdx.x;
    const size_t rbase = (size_t)row * N_COLS;

    // ---- CDNA5 async DMA: global -> LDS, b128 per lane, ASYNCcnt-tracked.
    //      th:TH_LOAD_NT: 288 MiB read-once stream, bypass-rinse the caches. ----
    {
        unsigned lds_t = (unsigned)(unsigned long long)(&sh_t[0]) + (unsigned)tid * 16u;
        unsigned lds_p = (unsigned)(unsigned long long)(&sh_p[0]) + (unsigned)tid * 16u;
        unsigned goff  = (unsigned)tid * 16u;
        unsigned long long gt = (unsigned long long)(targ + rbase);
        unsigned long long gp = (unsigned long long)(pred + rbase);
#pragma unroll
        for (int k = 0; k < F4_PER_THREAD; ++k) {
            asm volatile("global_load_async_to_lds_b128 %0, %1, %2 th:TH_LOAD_NT"
                         :: "v"(lds_t), "v"(goff), "s"(gt) : "memory");
            asm volatile("global_load_async_to_lds_b128 %0, %1, %2 th:TH_LOAD_NT"
                         :: "v"(lds_p), "v"(goff), "s"(gp) : "memory");
            lds_t += THREADS * 16u;
            lds_p += THREADS * 16u;
            goff  += THREADS * 16u;
        }
    }

    // ---- mask: direct non-temporal loads (overlap the DMA) ----
    unsigned mb[F4_PER_THREAD];
    {
        const unsigned* m = mask32 + (size_t)row * (N_COLS / 4);
#pragma unroll
        for (int k = 0; k < F4_PER_THREAD; ++k)
            mb[k] = __builtin_nontemporal_load(m + k * THREADS + tid);
    }

    // Drain this wave's async transfers, then rendezvous the workgroup.
    asm volatile("s_wait_asynccnt 0" ::: "memory");
    __syncthreads();

    // ---- pass 1: masked row maxima + valid count ----
    float tmax = -3.0e38f, pmax = -3.0e38f, nv = 0.0f;
#pragma unroll
    for (int k = 0; k < F4_PER_THREAD; ++k) {
        float4 t4 = sh_t[k * THREADS + tid];
        float4 p4 = sh_p[k * THREADS + tid];
        unsigned m = mb[k];
        if (m & 0x000000FFu) { tmax = fmaxf(tmax, t4.x); pmax = fmaxf(pmax, p4.x); nv += 1.0f; }
        if (m & 0x0000FF00u) { tmax = fmaxf(tmax, t4.y); pmax = fmaxf(pmax, p4.y); nv += 1.0f; }
        if (m & 0x00FF0000u) { tmax = fmaxf(tmax, t4.z); pmax = fmaxf(pmax, p4.z); nv += 1.0f; }
        if (m & 0xFF000000u) { tmax = fmaxf(tmax, t4.w); pmax = fmaxf(pmax, p4.w); nv += 1.0f; }
    }
    block_reduce3(tmax, pmax, nv, /*is_max_ab=*/true, sredA);

    // ---- pass 2: Zt = sum e^(t-tmax), Zp = sum e^(p-pmax), S = sum e^(t-tmax)*(p-pmax) ----
    float Zt = 0.0f, Zp = 0.0f, S = 0.0f;
#pragma unroll
    for (int k = 0; k < F4_PER_THREAD; ++k) {
        float4 t4 = sh_t[k * THREADS + tid];
        float4 p4 = sh_p[k * THREADS + tid];
        unsigned m = mb[k];
        if (m & 0x000000FFu) { float et = __expf(t4.x - tmax); float d = p4.x - pmax; Zt += et; Zp += __expf(d); S += et * d; }
        if (m & 0x0000FF00u) { float et = __expf(t4.y - tmax); float d = p4.y - pmax; Zt += et; Zp += __expf(d); S += et * d; }
        if (m & 0x00FF0000u) { float et = __expf(t4.z - tmax); float d = p4.z - pmax; Zt += et; Zp += __expf(d); S += et * d; }
        if (m & 0xFF000000u) { float et = __expf(t4.w - tmax); float d = p4.w - pmax; Zt += et; Zp += __expf(d); S += et * d; }
    }
    block_reduce3(Zt, Zp, S, /*is_max_ab=*/false, sredB);

    if (tid == 0) {
        // ce = -sum(q_i * logsoftmax(p)_i) = log(Zp) - S/Zt  (since sum q_i == 1)
        const bool ok = (nv >= 2.0f);
        ce_out[row] = ok ? (__logf(Zp) - S / Zt) : 0.0f;
        ok_out[row] = ok ? 1.0f : 0.0f;
    }
}

__global__ __launch_bounds__(256)
void listnet_final_reduce(const float* __restrict__ ce,
                          const float* __restrict__ ok,
                          float* __restrict__ out, int B)
{
    __shared__ float s1[256];
    __shared__ float s2[256];
    const int tid = threadIdx.x;
    float a = 0.0f, b = 0.0f;
    for (int i = tid; i < B; i += 256) { a += ce[i]; b += ok[i]; }
    s1[tid] = a; s2[tid] = b;
    __syncthreads();
#pragma unroll
    for (int s = 128; s > 0; s >>= 1) {
        if (tid < s) { s1[tid] += s1[tid + s]; s2[tid] += s2[tid + s]; }
        __syncthreads();
    }
    if (tid == 0) out[0] = (s2[0] > 0.0f) ? (s1[0] / s2[0]) : 0.0f;
}

extern "C" void kernel_launch(void* const* d_in, const int* in_sizes, int n_in,
                              void* d_out, int out_size, void* d_ws, size_t ws_size,
                              hipStream_t stream)
{
    const float*    pred   = (const float*)d_in[0];
    const float*    targ   = (const float*)d_in[1];
    const unsigned* mask32 = (const unsigned*)d_in[2];  // jnp.bool_ = 1 byte/elem, read 4 at a time

    const int B = in_sizes[0] / N_COLS;   // 4096

    float* ce = (float*)d_ws;             // B floats
    float* ok = ce + B;                   // B floats

    listnet_row_kernel<<<dim3(B), dim3(THREADS), 0, stream>>>(pred, targ, mask32, ce, ok);
    listnet_final_reduce<<<dim3(1), dim3(256), 0, stream>>>(ce, ok, (float*)d_out, B);
}